// NLMSFilter_3659312136401
// MI455X (gfx1250) — compile-verified
//
#include <hip/hip_runtime.h>
#include <cstdint>
#include <cstddef>

// ---------------- problem constants (from the reference) ----------------
#define NB     8
#define NK     513
#define NM     4096
#define NL     15                 // filter order
#define NTAU   3                  // delay
#define NSTART (NTAU + NL - 1)    // 17: first adapted frame
#define NSEQ   (NB * NK)          // 4104 independent sequences
#define EPSV   1e-8f

// ---------------- kernel tiling ----------------
#define BLOCK 128                 // sequences per block (1 lane = 1 sequence)
#define FR    16                  // frames per LDS tile
#define NT    (NM / FR)           // 256 tiles
static constexpr size_t PLANE = (size_t)NSEQ * (size_t)NM;

typedef float v4f __attribute__((ext_vector_type(4)));

// ---------------- CDNA5 async global->LDS copy (ASYNCcnt path) ----------------
typedef __attribute__((address_space(1))) int as1_int;   // global
typedef __attribute__((address_space(3))) int as3_int;   // LDS

__device__ __forceinline__ void async_g2l_b32(const float* g, const float* l) {
#if __has_builtin(__builtin_amdgcn_global_load_async_to_lds_b32)
  // (global int*, lds int*, imm offset, imm cpol) -> global_load_async_to_lds_b32
  __builtin_amdgcn_global_load_async_to_lds_b32(
      (as1_int*)(uintptr_t)g,
      (as3_int*)(uintptr_t)(uint32_t)(uintptr_t)l,
      0, 0);
#else
  // GV mode: VDST = LDS byte address, VADDR = 64-bit global address
  uint32_t lds_addr = (uint32_t)(uintptr_t)l;
  asm volatile("global_load_async_to_lds_b32 %0, %1, off"
               :: "v"(lds_addr), "v"(g) : "memory");
#endif
}

__device__ __forceinline__ void wait_async0() {
#if __has_builtin(__builtin_amdgcn_s_wait_asynccnt)
  __builtin_amdgcn_s_wait_asynccnt(0);
#else
  asm volatile("s_wait_asynccnt 0" ::: "memory");
#endif
}

// ---------------- main NLMS kernel ----------------
// One lane per (b,k) sequence. 15 complex taps + 18-deep complex delay line
// live in VGPRs; Y tiles are DMA'd into LDS (double-buffered) for coalescing.
__global__ __launch_bounds__(BLOCK)
void nlms_kernel(const float* __restrict__ Yre,
                 const float* __restrict__ Yim,
                 const float* __restrict__ mu_logit,
                 float* __restrict__ out)
{
  // frame-major LDS tiles: [buffer][frame*BLOCK + seqLocal]  (conflict-free reads)
  __shared__ float sRe[2][FR * BLOCK];
  __shared__ float sIm[2][FR * BLOCK];

  const int  tid     = threadIdx.x;
  const int  seqBase = blockIdx.x * BLOCK;
  const int  seq     = seqBase + tid;
  const bool active  = (seq < NSEQ);

  // mu = sigmoid(mu_logit)  (scalar, broadcast via constant cache)
  const float mu = 1.0f / (1.0f + __expf(-mu_logit[0]));

  // register-resident filter state
  float hre[NL], him[NL];
  float dre[NSTART + 1], dim_[NSTART + 1];   // d[i] = Y[m - i], i = 0..17
#pragma unroll
  for (int i = 0; i < NL; ++i) { hre[i] = 0.0f; him[i] = 0.0f; }
#pragma unroll
  for (int i = 0; i <= NSTART; ++i) { dre[i] = 0.0f; dim_[i] = 0.0f; }

  float* outEr = out;
  float* outEi = out + PLANE;
  float* outZr = out + 2 * PLANE;
  float* outZi = out + 3 * PLANE;
  const size_t obase = (size_t)seq * NM;

  // cooperative coalesced tile loader: BLOCK seqs x FR frames x {re,im}
  // branch-free: pad lanes clamp to a valid sequence (duplicate load, never read)
  auto issue_tile = [&](int t, int buf) {
    const int m0 = t * FR;
#pragma unroll
    for (int it = 0; it < FR; ++it) {
      const int g   = it * BLOCK + tid;  // consecutive tids -> consecutive addrs
      const int sl  = g >> 4;            // g / FR
      const int j   = g & (FR - 1);      // g % FR
      const int sg  = seqBase + sl;
      const int sgc = (sg < NSEQ) ? sg : (NSEQ - 1);   // clamp, no exec churn
      const size_t goff = (size_t)sgc * NM + (size_t)(m0 + j);
      async_g2l_b32(Yre + goff, &sRe[buf][j * BLOCK + sl]);
      async_g2l_b32(Yim + goff, &sIm[buf][j * BLOCK + sl]);
    }
  };

  // prime the pipeline
  issue_tile(0, 0);
  wait_async0();
  __syncthreads();

  for (int t = 0; t < NT; ++t) {
    const int cur = t & 1;
    if (t + 1 < NT) issue_tile(t + 1, cur ^ 1);   // overlap DMA with compute

    const int m0 = t * FR;
#pragma unroll
    for (int j4 = 0; j4 < FR; j4 += 4) {
      // 4-frame register output buffers -> one B128 NT store per stream
      v4f vEr, vEi, vZr, vZi;
#pragma unroll
      for (int jj = 0; jj < 4; ++jj) {
        const int j = j4 + jj;
        // shift delay line (fully renamed inside the unrolled tile body)
#pragma unroll
        for (int i = NSTART; i > 0; --i) { dre[i] = dre[i - 1]; dim_[i] = dim_[i - 1]; }
        dre[0]  = sRe[cur][j * BLOCK + tid];
        dim_[0] = sIm[cur][j * BLOCK + tid];

        const int m = m0 + j;
        float er, ei, zr, zi;
        if (m >= NSTART) {
          // z = sum_l conj(h_l) * w_l ; power = sum |w_l|^2 + eps  (4-way trees)
          float zra[4] = {0, 0, 0, 0}, zia[4] = {0, 0, 0, 0};
          float pa[4]  = {EPSV, 0, 0, 0};
#pragma unroll
          for (int l = 0; l < NL; ++l) {
            const float wr = dre[NTAU + l], wi = dim_[NTAU + l];
            const float hr = hre[l],        hi = him[l];
            zra[l & 3] = fmaf(hr, wr, fmaf( hi, wi, zra[l & 3]));
            zia[l & 3] = fmaf(hr, wi, fmaf(-hi, wr, zia[l & 3]));
            pa[l & 3]  = fmaf(wr, wr, fmaf( wi, wi, pa[l & 3]));
          }
          zr = (zra[0] + zra[1]) + (zra[2] + zra[3]);
          zi = (zia[0] + zia[1]) + (zia[2] + zia[3]);
          const float pw = (pa[0] + pa[1]) + (pa[2] + pa[3]);

          er = dre[0] - zr;
          ei = dim_[0] - zi;

          // s = mu / power  (v_rcp_f32 keeps the serial chain short)
          const float s   = mu * __builtin_amdgcn_rcpf(pw);
          const float ser = s * er, sei = s * ei;

          // h += s * conj(e) * w
#pragma unroll
          for (int l = 0; l < NL; ++l) {
            const float wr = dre[NTAU + l], wi = dim_[NTAU + l];
            hre[l] = fmaf(ser, wr, fmaf( sei, wi, hre[l]));
            him[l] = fmaf(ser, wi, fmaf(-sei, wr, him[l]));
          }
        } else {
          // passthrough region: E = Y, Z = 0 (delay line still advances)
          zr = 0.0f; zi = 0.0f; er = dre[0]; ei = dim_[0];
        }
        vEr[jj] = er; vEi[jj] = ei; vZr[jj] = zr; vZi[jj] = zi;
      }

      if (active) {
        // streaming output, no reuse -> 16B non-temporal stores keep the
        // 134MB input resident in the 192MB L2 while outputs bypass it
        const size_t o = obase + (size_t)(m0 + j4);
        __builtin_nontemporal_store(vEr, (v4f*)(outEr + o));
        __builtin_nontemporal_store(vEi, (v4f*)(outEi + o));
        __builtin_nontemporal_store(vZr, (v4f*)(outZr + o));
        __builtin_nontemporal_store(vZi, (v4f*)(outZi + o));
      }
    }

    wait_async0();     // next tile's DMA must have landed
    __syncthreads();   // ...and be visible to all lanes before we read it
  }
}

// ---------------- host entry ----------------
extern "C" void kernel_launch(void* const* d_in, const int* in_sizes, int n_in,
                              void* d_out, int out_size, void* d_ws, size_t ws_size,
                              hipStream_t stream) {
  (void)in_sizes; (void)n_in; (void)out_size; (void)d_ws; (void)ws_size;
  const float* Yre = (const float*)d_in[0];
  const float* Yim = (const float*)d_in[1];
  const float* mul = (const float*)d_in[2];
  float*       out = (float*)d_out;

  dim3 grid((NSEQ + BLOCK - 1) / BLOCK);   // 33 blocks x 128 lanes = 4224 >= 4104
  hipLaunchKernelGGL(nlms_kernel, grid, dim3(BLOCK), 0, stream, Yre, Yim, mul, out);
}